// MockMoETransformer_42296837931084
// MI455X (gfx1250) — compile-verified
//
#include <hip/hip_runtime.h>

// Problem constants (match reference)
#define Tt   1024     // B*S tokens
#define Hh   1024
#define Ii   4096
#define Ee   8
#define Ss   512
#define NHh  8
#define Vv   32000
#define Ll   2

typedef long long i64;
typedef float v2f __attribute__((ext_vector_type(2)));
typedef float v8f __attribute__((ext_vector_type(8)));

#define BM 16
#define BN 64
#define BK 64

#define GM_STORE        0
#define GM_BIAS_STORE   1
#define GM_BIAS_RES     2
#define GM_ACCUM_SCALED 3
#define GM_SILU_GATEUP  4
#define GM_NN_STORE     5

// C[M,N] = A[M,K] * op(B) (+ epilogue). Default op(B)=B^T with B stored [N,K]
// (weights are [out,in] row-major, K contiguous -> coalesced loads).
// GM_NN_STORE: B stored [K,N] (used for P*V).
// GM_SILU_GATEUP: two B tiles (gate rows n, up rows I+n); C = silu(g)*u.
// GM_ACCUM_SCALED: C[idx] += p1[m*p1s] * acc   (per-row expert weight).
// GM_BIAS_RES: C = acc + bias[n] + p1[idx]     (residual add).
template<int MODE>
__global__ __launch_bounds__(128)
void gemm_kernel(const float* __restrict__ A, const float* __restrict__ B,
                 const float* __restrict__ bias, const float* __restrict__ p1,
                 int p1s, i64 bOff2, float* __restrict__ C,
                 int M, int N, int K, int lda, int ldb, int ldc,
                 i64 aB, i64 aH, i64 bB, i64 bH, i64 cB, i64 cH, int nh)
{
  if (nh > 0) {                       // batched over (b, head)
    int z = blockIdx.z;
    int bb = z / nh, hh = z - bb * nh;
    A += bb * aB + hh * aH;
    B += bb * bB + hh * bH;
    C += bb * cB + hh * cH;
  }
  const int m0   = blockIdx.y * BM;
  const int n0   = blockIdx.x * BN;
  const int tid  = threadIdx.x;
  const int lane = tid & 31;
  const int wave = tid >> 5;

  __shared__ float As[BM][BK + 2];
  __shared__ float Bs[BN][BK + 2];
  __shared__ float Bs2[(MODE == GM_SILU_GATEUP) ? BN : 1][BK + 2];

  v8f acc  = {0.f, 0.f, 0.f, 0.f, 0.f, 0.f, 0.f, 0.f};
  v8f acc2 = {0.f, 0.f, 0.f, 0.f, 0.f, 0.f, 0.f, 0.f};

  // WMMA f32 16x16x4 lane mapping (ISA 7.12.2):
  // A: lane&15 = M row, (lane>>4)*2 = K-pair base (VGPR0/1 = k, k+1)
  // B: lane&15 = N col, (lane>>4)*2 = K row pair
  const int mrow  = lane & 15;
  const int khalf = (lane >> 4) << 1;
  const int nw    = wave << 4;

  const int r = tid >> 4;            // 0..7
  const int c = (tid & 15) << 2;     // 0..60, float4 granularity

  for (int k0 = 0; k0 < K; k0 += BK) {
    for (int rr = r; rr < BM; rr += 8) {
      const float4 v = *(const float4*)(A + (i64)(m0 + rr) * lda + (k0 + c));
      As[rr][c] = v.x; As[rr][c+1] = v.y; As[rr][c+2] = v.z; As[rr][c+3] = v.w;
    }
    if constexpr (MODE == GM_NN_STORE) {
      for (int rr = r; rr < BK; rr += 8) {           // rows are K, cols are N
        const float4 v = *(const float4*)(B + (i64)(k0 + rr) * ldb + (n0 + c));
        Bs[rr][c] = v.x; Bs[rr][c+1] = v.y; Bs[rr][c+2] = v.z; Bs[rr][c+3] = v.w;
      }
    } else {
      for (int rr = r; rr < BN; rr += 8) {           // rows are N, cols are K
        const float* bp = B + (i64)(n0 + rr) * ldb + (k0 + c);
        const float4 v = *(const float4*)bp;
        Bs[rr][c] = v.x; Bs[rr][c+1] = v.y; Bs[rr][c+2] = v.z; Bs[rr][c+3] = v.w;
        if (k0 + BK < K) __builtin_prefetch(bp + BK, 0, 1);  // global_prefetch next K tile
        if constexpr (MODE == GM_SILU_GATEUP) {
          const float4 u = *(const float4*)(B + bOff2 + (i64)(n0 + rr) * ldb + (k0 + c));
          Bs2[rr][c] = u.x; Bs2[rr][c+1] = u.y; Bs2[rr][c+2] = u.z; Bs2[rr][c+3] = u.w;
        }
      }
    }
    __syncthreads();

#pragma unroll
    for (int kk = 0; kk < BK; kk += 4) {
      v2f a, b;
      a.x = As[mrow][kk + khalf];
      a.y = As[mrow][kk + khalf + 1];
      if constexpr (MODE == GM_NN_STORE) {
        b.x = Bs[kk + khalf][nw + mrow];
        b.y = Bs[kk + khalf + 1][nw + mrow];
      } else {
        b.x = Bs[nw + mrow][kk + khalf];
        b.y = Bs[nw + mrow][kk + khalf + 1];
      }
      acc = __builtin_amdgcn_wmma_f32_16x16x4_f32(false, a, false, b,
                                                  (short)0, acc, false, false);
      if constexpr (MODE == GM_SILU_GATEUP) {
        v2f b2;
        b2.x = Bs2[nw + mrow][kk + khalf];
        b2.y = Bs2[nw + mrow][kk + khalf + 1];
        acc2 = __builtin_amdgcn_wmma_f32_16x16x4_f32(false, a, false, b2,
                                                     (short)0, acc2, false, false);
      }
    }
    __syncthreads();
  }

  // C layout: acc[rr] -> row m0 + rr + 8*(lane>=16), col n0 + nw + (lane&15)
  const int cn = n0 + nw + mrow;
  const int mb = m0 + ((lane >> 4) << 3);
#pragma unroll
  for (int rr = 0; rr < 8; ++rr) {
    float v = acc[rr];
    if constexpr (MODE == GM_SILU_GATEUP) {
      const float g = acc[rr];
      v = (g / (1.f + __expf(-g))) * acc2[rr];
    }
    const i64 idx = (i64)(mb + rr) * ldc + cn;
    if constexpr (MODE == GM_BIAS_STORE) v += bias[cn];
    if constexpr (MODE == GM_BIAS_RES)   v += bias[cn] + p1[idx];
    if constexpr (MODE == GM_ACCUM_SCALED) {
      C[idx] += p1[(i64)(mb + rr) * p1s] * v;
    } else {
      C[idx] = v;
    }
  }
}

__global__ __launch_bounds__(256)
void layernorm_kernel(const float* __restrict__ x, const float* __restrict__ scale,
                      const float* __restrict__ bias, float* __restrict__ out, int H)
{
  const int t = blockIdx.x;
  const float* row = x + (i64)t * H;
  float* orow = out + (i64)t * H;
  __shared__ float red[256];
  const int tid = threadIdx.x;

  float s = 0.f;
  for (int i = tid; i < H; i += 256) s += row[i];
  red[tid] = s; __syncthreads();
  for (int o = 128; o > 0; o >>= 1) { if (tid < o) red[tid] += red[tid + o]; __syncthreads(); }
  const float mean = red[0] / (float)H;
  __syncthreads();

  float v = 0.f;
  for (int i = tid; i < H; i += 256) { const float d = row[i] - mean; v += d * d; }
  red[tid] = v; __syncthreads();
  for (int o = 128; o > 0; o >>= 1) { if (tid < o) red[tid] += red[tid + o]; __syncthreads(); }
  const float inv = rsqrtf(red[0] / (float)H + 1e-5f);

  for (int i = tid; i < H; i += 256)
    orow[i] = (row[i] - mean) * inv * scale[i] + bias[i];
}

__global__ __launch_bounds__(128)
void softmax_kernel(float* __restrict__ sc, int S, float scale)
{
  float* p = sc + (i64)blockIdx.x * S;
  __shared__ float red[128];
  const int tid = threadIdx.x;

  float mx = -3.0e38f;
  for (int i = tid; i < S; i += 128) mx = fmaxf(mx, p[i] * scale);
  red[tid] = mx; __syncthreads();
  for (int o = 64; o > 0; o >>= 1) { if (tid < o) red[tid] = fmaxf(red[tid], red[tid + o]); __syncthreads(); }
  mx = red[0]; __syncthreads();

  float sum = 0.f;
  for (int i = tid; i < S; i += 128) { const float e = __expf(p[i] * scale - mx); p[i] = e; sum += e; }
  red[tid] = sum; __syncthreads();
  for (int o = 64; o > 0; o >>= 1) { if (tid < o) red[tid] += red[tid + o]; __syncthreads(); }
  const float rcp = 1.f / red[0];
  for (int i = tid; i < S; i += 128) p[i] *= rcp;
}

// logits[t][e] = dot(hln[t], gate_w[e]); one wave per expert, block per token.
__global__ __launch_bounds__(256)
void gate_kernel(const float* __restrict__ hln, const float* __restrict__ gw,
                 float* __restrict__ logits, int H)
{
  const int t = blockIdx.x;
  const int e = threadIdx.x >> 5;
  const int lane = threadIdx.x & 31;
  const float* row = hln + (i64)t * H;
  const float* w = gw + (i64)e * H;
  float s = 0.f;
  for (int i = lane; i < H; i += 32) s += row[i] * w[i];
  for (int o = 16; o > 0; o >>= 1) s += __shfl_xor(s, o, 32);
  if (lane == 0) logits[t * Ee + e] = s;
}

// softmax over E, top-2, renormalize -> dense combine weights cw[T,E]
__global__ __launch_bounds__(64)
void router_kernel(const float* __restrict__ logits, float* __restrict__ cw, int T)
{
  const int t = blockIdx.x * 64 + threadIdx.x;
  if (t >= T) return;
  float l[Ee];
  float mx = -3.0e38f;
  for (int e = 0; e < Ee; ++e) { l[e] = logits[t * Ee + e]; mx = fmaxf(mx, l[e]); }
  float sum = 0.f;
  for (int e = 0; e < Ee; ++e) { l[e] = __expf(l[e] - mx); sum += l[e]; }
  const float r = 1.f / sum;
  for (int e = 0; e < Ee; ++e) l[e] *= r;
  int i1 = 0; float v1 = -1.f;
  for (int e = 0; e < Ee; ++e) if (l[e] > v1) { v1 = l[e]; i1 = e; }
  int i2 = -1; float v2 = -1.f;
  for (int e = 0; e < Ee; ++e) if (e != i1 && l[e] > v2) { v2 = l[e]; i2 = e; }
  const float tot = 1.f / (v1 + v2);
  for (int e = 0; e < Ee; ++e) cw[t * Ee + e] = 0.f;
  cw[t * Ee + i1] = v1 * tot;
  cw[t * Ee + i2] = v2 * tot;
}

extern "C" void kernel_launch(void* const* d_in, const int* in_sizes, int n_in,
                              void* d_out, int out_size, void* d_ws, size_t ws_size,
                              hipStream_t stream)
{
  (void)in_sizes; (void)n_in; (void)out_size; (void)ws_size;

  const float* x_in  = (const float*)d_in[0];
  const float* ln1_s = (const float*)d_in[1];
  const float* ln1_b = (const float*)d_in[2];
  const float* in_w  = (const float*)d_in[3];
  const float* in_b  = (const float*)d_in[4];
  const float* out_w = (const float*)d_in[5];
  const float* out_b = (const float*)d_in[6];
  const float* ln2_s = (const float*)d_in[7];
  const float* ln2_b = (const float*)d_in[8];
  const float* gw    = (const float*)d_in[9];
  const float* guw   = (const float*)d_in[10];
  const float* dnw   = (const float*)d_in[11];
  const float* lm_w  = (const float*)d_in[12];
  const float* lm_b  = (const float*)d_in[13];
  float* out = (float*)d_out;

  float* ws   = (float*)d_ws;
  float* hln  = ws;  ws += (i64)Tt * Hh;          // 1M
  float* qkv  = ws;  ws += (i64)Tt * 3 * Hh;      // 3M
  float* big  = ws;  ws += (i64)16 * Ss * Ss;     // 4M (scores / expert hidden, == T*I)
  float* obuf = ws;  ws += (i64)Tt * Hh;          // 1M
  float* cur  = ws;  ws += (i64)Tt * Hh;          // 1M
  float* glog = ws;  ws += (i64)Tt * Ee;
  float* cwb  = ws;  ws += (i64)Tt * Ee;

  const float iscale = 0.08838834764831845f;      // 1/sqrt(128)
  const i64 Z0 = 0;

  for (int i = 0; i < Ll; ++i) {
    const float* src = (i == 0) ? x_in : cur;

    // LN1
    layernorm_kernel<<<Tt, 256, 0, stream>>>(src, ln1_s + i * Hh, ln1_b + i * Hh, hln, Hh);

    // qkv = hln * in_w^T + in_b : [T,3H]
    gemm_kernel<GM_BIAS_STORE><<<dim3(3 * Hh / BN, Tt / BM, 1), 128, 0, stream>>>(
        hln, in_w + (i64)i * 3 * Hh * Hh, in_b + (i64)i * 3 * Hh, nullptr, 0, Z0,
        qkv, Tt, 3 * Hh, Hh, Hh, Hh, 3 * Hh, Z0, Z0, Z0, Z0, Z0, Z0, 0);

    // scores[b,h] = Q * K^T : batched over 16 (b,h)
    gemm_kernel<GM_STORE><<<dim3(Ss / BN, Ss / BM, 2 * NHh), 128, 0, stream>>>(
        qkv, qkv + Hh, nullptr, nullptr, 0, Z0, big,
        Ss, Ss, 128, 3 * Hh, 3 * Hh, Ss,
        (i64)Ss * 3 * Hh, 128, (i64)Ss * 3 * Hh, 128,
        (i64)NHh * Ss * Ss, (i64)Ss * Ss, NHh);

    // softmax rows with 1/sqrt(d)
    softmax_kernel<<<2 * NHh * Ss, 128, 0, stream>>>(big, Ss, iscale);

    // O[b,h] = P * V  (B stored [K,N])
    gemm_kernel<GM_NN_STORE><<<dim3(128 / BN, Ss / BM, 2 * NHh), 128, 0, stream>>>(
        big, qkv + 2 * Hh, nullptr, nullptr, 0, Z0, obuf,
        Ss, 128, Ss, Ss, 3 * Hh, Hh,
        (i64)NHh * Ss * Ss, (i64)Ss * Ss, (i64)Ss * 3 * Hh, 128,
        (i64)Ss * Hh, 128, NHh);

    // cur = obuf * out_w^T + out_b + residual
    gemm_kernel<GM_BIAS_RES><<<dim3(Hh / BN, Tt / BM, 1), 128, 0, stream>>>(
        obuf, out_w + (i64)i * Hh * Hh, out_b + (i64)i * Hh, src, 0, Z0,
        cur, Tt, Hh, Hh, Hh, Hh, Hh, Z0, Z0, Z0, Z0, Z0, Z0, 0);

    // LN2
    layernorm_kernel<<<Tt, 256, 0, stream>>>(cur, ln2_s + i * Hh, ln2_b + i * Hh, hln, Hh);

    // router
    gate_kernel<<<Tt, 256, 0, stream>>>(hln, gw + (i64)i * Ee * Hh, glog, Hh);
    router_kernel<<<Tt / 64, 64, 0, stream>>>(glog, cwb, Tt);

    // dense experts (cw==0 for unselected -> exact); cur += cw[:,e] * expert_e(hln)
    for (int e = 0; e < Ee; ++e) {
      const float* we = guw + ((i64)i * Ee + e) * 2 * Ii * Hh;
      gemm_kernel<GM_SILU_GATEUP><<<dim3(Ii / BN, Tt / BM, 1), 128, 0, stream>>>(
          hln, we, nullptr, nullptr, 0, (i64)Ii * Hh,
          big, Tt, Ii, Hh, Hh, Hh, Ii, Z0, Z0, Z0, Z0, Z0, Z0, 0);
      const float* wd = dnw + ((i64)i * Ee + e) * Hh * Ii;
      gemm_kernel<GM_ACCUM_SCALED><<<dim3(Hh / BN, Tt / BM, 1), 128, 0, stream>>>(
          big, wd, nullptr, cwb + e, Ee, Z0,
          cur, Tt, Hh, Ii, Ii, Ii, Hh, Z0, Z0, Z0, Z0, Z0, Z0, 0);
    }
  }

  // logits = cur * lm_w^T + lm_b : [T,V]
  gemm_kernel<GM_BIAS_STORE><<<dim3(Vv / BN, Tt / BM, 1), 128, 0, stream>>>(
      cur, lm_w, lm_b, nullptr, 0, Z0,
      out, Tt, Vv, Hh, Hh, Hh, Vv, Z0, Z0, Z0, Z0, Z0, Z0, 0);
}